// CayleyConv_61409442399018
// MI455X (gfx1250) — compile-verified
//
#include <hip/hip_runtime.h>

// CayleyConv for MI455X (gfx1250, wave32).
// Key algebraic optimization: the output symmetrization 0.5*(t=3i+j <-> 3j+i)
// commutes with both the channel-mix GEMM and the tap fold, so it is pushed
// into the W tensor: only 6 unique symmetric components are computed through
// the whole pipeline (fold VALU -25%, folded intermediate 28.4MB vs 42.6MB,
// GEMM 1.5x fewer v_wmma ops). Expansion 6->9 is a compile-time store pattern
// at output write (no LUT, no extra loads).
// Pipeline: prep (Cayley G -> Wsym[6] + c_mat^2) -> fold (tap gather, x is
// 38MB -> L2-resident) -> fused fp32 WMMA GEMM + 6->9 expand to d_out.
// GEMM is exact fp32 via v_wmma_f32_16x16x4_f32 (K=128 = 32 k-steps).
// Workspace requirement: ~29 MB.

typedef float v2f __attribute__((ext_vector_type(2)));
typedef float v8f __attribute__((ext_vector_type(8)));

#define Bn    8
#define CINc  128
#define COUTc 128
#define Hn    32
#define Wn    32
#define HOo   34
#define WOo   34
#define SP    (HOo * WOo)       // 1156
#define USYM  6                 // unique symmetric t-components
#define MTOT  (Bn * SP * USYM)  // 55488 = 578 * 96 exactly
#define NSUP  (MTOT / 96)       // 578 supertiles of 96 m-columns
#define EPSF  1e-7f

// float-offsets into workspace
#define WS_W    0               // 486 floats: Wsym[u][ij][pq]
#define WS_CM   1024            // 16384 floats: c_mat[c][o]
#define WS_FOLD (1024 + 16384)  // CIN*MTOT floats: foldedS[c][m], m=(b,s,u)

// ---------------------------------------------------------------- prep ----
__global__ void prep_kernel(const float* __restrict__ g, float* __restrict__ ws) {
  __shared__ float Gs[9][3][3];
  int tid = threadIdx.x;
  if (tid < 9) {
    const int nums[9] = {0, 1, 2, 3, 0, 5, 6, 7, 1};
    float a = g[2 * nums[tid]];
    float c = g[2 * nums[tid] + 1];
    float f00 = 1.f, f01 = -a, f02 = -a;
    float f10 = a,   f11 = 1.f, f12 = -c;
    float f20 = a,   f21 = c,   f22 = 1.f;
    if (tid == 4) { f01 = f02 = f10 = f12 = f20 = f21 = 0.f; }
    float det = f00 * (f11 * f22 - f12 * f21)
              - f01 * (f10 * f22 - f12 * f20)
              + f02 * (f10 * f21 - f11 * f20) + EPSF;
    float id = 1.f / det;
    float i00 = (f11 * f22 - f12 * f21) * id, i01 = (f02 * f21 - f01 * f22) * id,
          i02 = (f01 * f12 - f02 * f11) * id;
    float i10 = (f12 * f20 - f10 * f22) * id, i11 = (f00 * f22 - f02 * f20) * id,
          i12 = (f02 * f10 - f00 * f12) * id;
    float i20 = (f10 * f21 - f11 * f20) * id, i21 = (f01 * f20 - f00 * f21) * id,
          i22 = (f00 * f11 - f01 * f10) * id;
    float s00 = 2.f - f00, s01 = -f01, s02 = -f02;
    float s10 = -f10, s11 = 2.f - f11, s12 = -f12;
    float s20 = -f20, s21 = -f21, s22 = 2.f - f22;
    Gs[tid][0][0] = i00 * s00 + i01 * s10 + i02 * s20;
    Gs[tid][0][1] = i00 * s01 + i01 * s11 + i02 * s21;
    Gs[tid][0][2] = i00 * s02 + i01 * s12 + i02 * s22;
    Gs[tid][1][0] = i10 * s00 + i11 * s10 + i12 * s20;
    Gs[tid][1][1] = i10 * s01 + i11 * s11 + i12 * s21;
    Gs[tid][1][2] = i10 * s02 + i11 * s12 + i12 * s22;
    Gs[tid][2][0] = i20 * s00 + i21 * s10 + i22 * s20;
    Gs[tid][2][1] = i20 * s01 + i21 * s11 + i22 * s21;
    Gs[tid][2][2] = i20 * s02 + i21 * s12 + i22 * s22;
  }
  __syncthreads();
  // Wsym[u][(mi,mj)][(p,q)]: u -> (t1,t2) pairs; off-diag averaged with 0.5
  if (tid < USYM) {
    const int t1s[USYM] = {0, 1, 2, 4, 5, 8};
    const int t2s[USYM] = {0, 3, 6, 4, 7, 8};
    int t1 = t1s[tid], t2 = t2s[tid];
    #pragma unroll
    for (int mi = 0; mi < 3; ++mi)
      #pragma unroll
      for (int mj = 0; mj < 3; ++mj)
        #pragma unroll
        for (int p = 0; p < 3; ++p)
          #pragma unroll
          for (int q = 0; q < 3; ++q) {
            float w1 = Gs[t1][mi][p] * Gs[t1][mj][q];
            float w2 = Gs[t2][mi][p] * Gs[t2][mj][q];
            ws[WS_W + tid * 81 + (mi * 3 + mj) * 9 + (p * 3 + q)] =
                (t1 == t2) ? w1 : 0.5f * (w1 + w2);
          }
  }
  for (int i = tid; i < CINc * COUTc; i += blockDim.x) {
    float v = g[16 + i];
    ws[WS_CM + i] = v * v;
  }
}

// ---------------------------------------------------------------- fold ----
// gid = ((c*B + b)*SP + s). Adjacent threads -> adjacent s -> coalesced x
// reads; u-innermost layout makes each thread's 6 stores CONTIGUOUS.
__global__ void fold_kernel(const float* __restrict__ x,
                            const float* __restrict__ ws,
                            float* __restrict__ fold) {
  __shared__ float Wl[USYM * 81];
  for (int i = threadIdx.x; i < USYM * 81; i += blockDim.x) Wl[i] = ws[WS_W + i];
  __syncthreads();

  int gid = blockIdx.x * blockDim.x + threadIdx.x;
  int s   = gid % SP;
  int rem = gid / SP;
  int b   = rem % Bn;
  int cch = rem / Bn;
  int ho = s / WOo, wo = s % WOo;

  float acc[USYM];
  #pragma unroll
  for (int u = 0; u < USYM; ++u) acc[u] = 0.f;

  #pragma unroll
  for (int mi = 0; mi < 3; ++mi) {
    int h = ho - mi;
    if (h < 0 || h >= Hn) continue;
    #pragma unroll
    for (int mj = 0; mj < 3; ++mj) {
      int w = wo - mj;
      if (w < 0 || w >= Wn) continue;
      const float* xp = x + (size_t)((((b * CINc + cch) * Hn + h) * Wn + w)) * 9;
      float X[9];
      #pragma unroll
      for (int k = 0; k < 9; ++k) X[k] = xp[k];
      int e = (mi * 3 + mj) * 9;
      #pragma unroll
      for (int u = 0; u < USYM; ++u) {
        float sum = 0.f;
        #pragma unroll
        for (int k = 0; k < 9; ++k) sum += Wl[u * 81 + e + k] * X[k];
        acc[u] += sum;
      }
    }
  }
  // m = (b*SP + s)*6 + u  (u-innermost -> contiguous 6-float store)
  size_t base = (size_t)cch * MTOT + (size_t)(b * SP + s) * USYM;
  #pragma unroll
  for (int u = 0; u < USYM; ++u) fold[base + u] = acc[u];
}

// ------------------------------------------------------- gemm + expand ----
// Block = 6 waves, D supertile 16(o) x 96(m) = 16 complete 6-u groups.
// Each wave: one 16x16 tile, 32 x v_wmma_f32_16x16x4_f32 over K=128.
// Fragment layout (per ISA 32-bit 16x4 A table, B mirrored):
//  A: lanes0-15 M=lane holds K=k,k+1; lanes16-31 M=lane-16 holds K=k+2,k+3
//  B: lanes0-15 N=lane holds K=k,k+1; lanes16-31 N=lane-16 holds K=k+2,k+3
//  D: VGPR r: lanes0-15 (M=r, N=lane); lanes16-31 (M=r+8, N=lane-16)
// D staged in LDS (row pad 97: 8*97 % 64banks = 8 -> no aliasing). Expansion
// 6->9 is a hardcoded store pattern: 6 LDS reads -> 9 contiguous stores.
// Grid exact (578 x 8), blocks fully active -> EXEC all-1s for WMMA.
__global__ void gemm_sym_kernel(const float* __restrict__ ws,
                                float* __restrict__ out) {
  __shared__ float Dl[16][97];
  const float* cm   = ws + WS_CM;   // c_mat[c][o]
  const float* fold = ws + WS_FOLD; // foldedS[c][m]
  int wave = threadIdx.x >> 5;      // 0..5
  int lane = threadIdx.x & 31;
  int o0 = blockIdx.y * 16;
  int m0 = blockIdx.x * 96 + wave * 16;
  int lh   = lane & 15;
  int koff = (lane >= 16) ? 2 : 0;

  v8f acc = {};
  #pragma unroll 4
  for (int k0 = 0; k0 < CINc; k0 += 4) {
    int kb = k0 + koff;
    v2f a, bf;
    a.x  = cm[(kb + 0) * COUTc + o0 + lh];
    a.y  = cm[(kb + 1) * COUTc + o0 + lh];
    bf.x = fold[(size_t)(kb + 0) * MTOT + m0 + lh];
    bf.y = fold[(size_t)(kb + 1) * MTOT + m0 + lh];
    acc = __builtin_amdgcn_wmma_f32_16x16x4_f32(false, a, false, bf,
                                                (short)0, acc, false, false);
  }

  int oh = (lane >= 16) ? 8 : 0;
  #pragma unroll
  for (int r = 0; r < 8; ++r) Dl[r + oh][wave * 16 + lh] = acc[r];
  __syncthreads();

  // Expand u(6) -> 3x3: one thread per (o_l, gl) group; 256 groups / 192 thr.
  // Stores are 9 contiguous floats; symmetric entries replicated (0.5
  // averaging already folded into Wsym).
  int g0 = blockIdx.x * 16;  // base (b,s) group index
  for (int grp = threadIdx.x; grp < 256; grp += 192) {
    int o_l = grp >> 4;   // 0..15
    int gl  = grp & 15;   // 0..15
    const float* dp = &Dl[o_l][gl * USYM];
    float u0 = dp[0], u1 = dp[1], u2 = dp[2], u3 = dp[3], u4 = dp[4], u5 = dp[5];
    int G = g0 + gl;
    int b = G / SP, s = G - b * SP;
    float* op = out + ((size_t)(b * COUTc + o0 + o_l) * SP + s) * 9;
    op[0] = u0; op[1] = u1; op[2] = u2;
    op[3] = u1; op[4] = u3; op[5] = u4;
    op[6] = u2; op[7] = u4; op[8] = u5;
  }
}

// -------------------------------------------------------------- launch ----
extern "C" void kernel_launch(void* const* d_in, const int* in_sizes, int n_in,
                              void* d_out, int out_size, void* d_ws, size_t ws_size,
                              hipStream_t stream) {
  const float* x = (const float*)d_in[0];  // (B, CIN, H, W, 3, 3) fp32
  const float* g = (const float*)d_in[1];  // 16 + CIN*COUT fp32
  float* ws  = (float*)d_ws;
  float* out = (float*)d_out;

  prep_kernel<<<1, 256, 0, stream>>>(g, ws);

  // CIN*B*SP = 1,183,744 = 4624 * 256 exactly
  fold_kernel<<<(CINc * Bn * SP) / 256, 256, 0, stream>>>(x, ws, ws + WS_FOLD);

  // 578 supertiles x 8 o-tiles, 6 waves (192 threads) per block, all exact
  dim3 gg(NSUP, COUTc / 16);
  gemm_sym_kernel<<<gg, 192, 0, stream>>>(ws, out);
}